// LossFunction_37005438222364
// MI455X (gfx1250) — compile-verified
//
#include <hip/hip_runtime.h>
#include <math.h>

typedef __attribute__((ext_vector_type(2))) float v2f;
typedef __attribute__((ext_vector_type(4))) float v4f;
typedef __attribute__((ext_vector_type(8))) float v8f;

#define NP 8
#define NG 4
#define NJ 14
#define BT 512
#define PDIM 42       // J*3
#define PPAD 44       // padded to multiple of 4 for K=4 WMMA steps
#define NPERM 1680    // 8*7*6*5
#define MSE_N (16*32*256*256)
#define MSE_BLOCKS 1024
#define MSE_THREADS 256
#define MSE_STRIDE (MSE_BLOCKS * MSE_THREADS)
#define MSE_ITERS ((MSE_N / 4) / MSE_STRIDE)   // 32, exact

// workspace layout (floats)
#define WS_MSE  0
#define WS_OFF  (WS_MSE + MSE_BLOCKS)
#define WS_SIZE (WS_OFF + BT)
#define WS_POSE (WS_SIZE + BT)

// ---------------- Kernel A: streaming MSE partial reduction ----------------
// 268 MB streamed exactly once (> 192 MB L2) -> non-temporal b128 loads so the
// stream does not thrash L2; 4 independent accumulators / 8 outstanding b128
// loads per unrolled body to cover HBM latency at 23.3 TB/s.
__global__ void __launch_bounds__(MSE_THREADS)
mse_partial_kernel(const float* __restrict__ a,
                   const float* __restrict__ b,
                   float* __restrict__ ws) {
    const int tid = blockIdx.x * MSE_THREADS + threadIdx.x;
    const v4f* a4 = (const v4f*)a;
    const v4f* b4 = (const v4f*)b;
    float s0 = 0.f, s1 = 0.f, s2 = 0.f, s3 = 0.f;
    #pragma unroll 4
    for (int it = 0; it < MSE_ITERS; ++it) {
        const int i = tid + it * MSE_STRIDE;
        v4f x = __builtin_nontemporal_load(&a4[i]);
        v4f y = __builtin_nontemporal_load(&b4[i]);
        float d0 = x.x - y.x, d1 = x.y - y.y, d2 = x.z - y.z, d3 = x.w - y.w;
        s0 += d0 * d0;
        s1 += d1 * d1;
        s2 += d2 * d2;
        s3 += d3 * d3;
    }
    float s = (s0 + s1) + (s2 + s3);
    // wave32 reduce
    #pragma unroll
    for (int off = 16; off > 0; off >>= 1) s += __shfl_xor(s, off, 32);
    __shared__ float red[MSE_THREADS / 32];
    const int lane = threadIdx.x & 31, wid = threadIdx.x >> 5;
    if (lane == 0) red[wid] = s;
    __syncthreads();
    if (threadIdx.x == 0) {
        float t = 0.f;
        #pragma unroll
        for (int w = 0; w < MSE_THREADS / 32; ++w) t += red[w];
        ws[WS_MSE + blockIdx.x] = t;
    }
}

// lexicographic decode of permutation index p in [0,1680): matches
// itertools.permutations(range(8), 4) ordering. Register-only (no dynamic
// array indexing -> predicated selects).
__device__ __forceinline__ void decode_perm(int p, int rows[NG]) {
    int cand[NP];
    #pragma unroll
    for (int m = 0; m < NP; ++m) cand[m] = m;
    const int divs[NG] = {210, 30, 5, 1};
    int rem = p;
    #pragma unroll
    for (int k = 0; k < NG; ++k) {
        int idx = rem / divs[k];
        rem -= idx * divs[k];
        int sel = 0;
        #pragma unroll
        for (int m = 0; m < NP; ++m) sel = (m == idx) ? cand[m] : sel;
        rows[k] = sel;
        // remove element idx (predicated shift keeps everything in VGPRs)
        #pragma unroll
        for (int m = 0; m < NP - 1; ++m) cand[m] = (m >= idx) ? cand[m + 1] : cand[m];
    }
}

// ---------------- Kernel B: Hungarian matching, one wave per (b,t) ----------
__global__ void __launch_bounds__(32)
hungarian_kernel(const float* __restrict__ hor_offset,
                 const float* __restrict__ hor_bsize,
                 const float* __restrict__ hor_center,
                 const float* __restrict__ scores,
                 const float* __restrict__ x_pose3d,
                 const float* __restrict__ gt_wh,
                 const float* __restrict__ gt_off,
                 const float* __restrict__ gt_center,
                 const float* __restrict__ gt_pose,
                 float* __restrict__ ws) {
    const int bt = blockIdx.x;
    const int lane = threadIdx.x; // 32 threads = one wave32

    __shared__ float ppL[NP][PPAD];
    __shared__ float gpL[NG][PPAD];
    __shared__ float Cbox[NP * NG];
    __shared__ float Cpose[NP * NG];
    __shared__ float npS[NP];
    __shared__ float ngS[NG];

    const float* pp = x_pose3d + (size_t)bt * NP * PDIM;
    const float* gp = gt_pose + (size_t)bt * NG * PDIM;

    // stage zero-padded pose matrices in LDS
    for (int idx = lane; idx < NP * PPAD; idx += 32) {
        int r = idx / PPAD, c = idx % PPAD;
        ppL[r][c] = (c < PDIM) ? pp[r * PDIM + c] : 0.f;
    }
    for (int idx = lane; idx < NG * PPAD; idx += 32) {
        int r = idx / PPAD, c = idx % PPAD;
        gpL[r][c] = (c < PDIM) ? gp[r * PDIM + c] : 0.f;
    }
    __syncthreads();

    // squared norms
    if (lane < NP) {
        float s = 0.f;
        for (int c = 0; c < PDIM; ++c) { float v = ppL[lane][c]; s += v * v; }
        npS[lane] = s;
    }
    if (lane < NG) {
        float s = 0.f;
        for (int c = 0; c < PDIM; ++c) { float v = gpL[lane][c]; s += v * v; }
        ngS[lane] = s;
    }
    // box cost: one entry per lane, C[i][j] = ||pc_i - gc_j|| - score_i
    {
        const int i = lane >> 2, j = lane & 3;
        const float* pc = hor_center + (size_t)bt * NP * 2;
        const float* gc = gt_center + (size_t)bt * NG * 2;
        const float* sc = scores + (size_t)bt * NP;
        float dx = pc[i * 2 + 0] - gc[j * 2 + 0];
        float dy = pc[i * 2 + 1] - gc[j * 2 + 1];
        Cbox[lane] = sqrtf(dx * dx + dy * dy) - sc[i];
    }
    __syncthreads();

    // ---- fp32 WMMA Gram matrix: G = Ppose(16x44, rows>=8 zero) x GposeT(44x16) ----
    // A layout (16x4 f32): lanes 0-15 -> M=lane, K={0,1}; lanes 16-31 -> M=lane-16, K={2,3}
    // B layout symmetric: lanes 0-15 -> N=lane, K={0,1}; lanes 16-31 -> N=lane-16, K={2,3}
    v8f acc = {};
    const int row = lane & 15;
    const bool hi = lane >= 16;
    const int ra = row & (NP - 1); // always in-bounds LDS reads
    const int rb = row & (NG - 1);
    #pragma unroll
    for (int s = 0; s < PPAD / 4; ++s) {
        const int k0 = 4 * s + (hi ? 2 : 0);
        const int k1 = 4 * s + (hi ? 3 : 1);
        v2f a, b;
        float ax = ppL[ra][k0], ay = ppL[ra][k1];
        float bx = gpL[rb][k0], by = gpL[rb][k1];
        a.x = (row < NP) ? ax : 0.f;
        a.y = (row < NP) ? ay : 0.f;
        b.x = (row < NG) ? bx : 0.f;
        b.y = (row < NG) ? by : 0.f;
        acc = __builtin_amdgcn_wmma_f32_16x16x4_f32(
            /*neg_a=*/false, a, /*neg_b=*/false, b,
            /*c_mod=*/(short)0, acc, /*reuse_a=*/false, /*reuse_b=*/false);
    }
    // C/D layout: VGPR i holds M=i (lanes 0-15, N=lane). Lanes 0..3 own column j.
    if (lane < NG) {
        #pragma unroll
        for (int i = 0; i < NP; ++i) {
            float c2 = npS[i] + ngS[lane] - 2.f * acc[i];
            Cpose[i * NG + lane] = sqrtf(fmaxf(c2, 0.f));
        }
    }
    __syncthreads();

    // ---- enumerate all 1680 permutations, dual argmin (box & pose) ----
    float bestB = 3.0e38f, bestP = 3.0e38f;
    int bpB = 0, bpP = 0;
    for (int p = lane; p < NPERM; p += 32) {
        int rows[NG];
        decode_perm(p, rows);
        float tb = Cbox[rows[0] * NG + 0] + Cbox[rows[1] * NG + 1] +
                   Cbox[rows[2] * NG + 2] + Cbox[rows[3] * NG + 3];
        float tp = Cpose[rows[0] * NG + 0] + Cpose[rows[1] * NG + 1] +
                   Cpose[rows[2] * NG + 2] + Cpose[rows[3] * NG + 3];
        if (tb < bestB) { bestB = tb; bpB = p; }
        if (tp < bestP) { bestP = tp; bpP = p; }
    }
    // cross-lane argmin; tie-break = lowest perm index (matches jnp.argmin)
    #pragma unroll
    for (int off = 16; off > 0; off >>= 1) {
        float ovB = __shfl_xor(bestB, off, 32);
        int opB = __shfl_xor(bpB, off, 32);
        if (ovB < bestB || (ovB == bestB && opB < bpB)) { bestB = ovB; bpB = opB; }
        float ovP = __shfl_xor(bestP, off, 32);
        int opP = __shfl_xor(bpP, off, 32);
        if (ovP < bestP || (ovP == bestP && opP < bpP)) { bestP = ovP; bpP = opP; }
    }

    if (lane == 0) {
        int rB[NG], rP[NG];
        decode_perm(bpB, rB);
        decode_perm(bpP, rP);
        const float* po = hor_offset + (size_t)bt * NP * 2;
        const float* go = gt_off + (size_t)bt * NG * 2;
        const float* psz = hor_bsize + (size_t)bt * NP * 4;
        const float* gsz = gt_wh + (size_t)bt * NG * 4;
        float off_l = 0.f, size_l = 0.f;
        #pragma unroll
        for (int j = 0; j < NG; ++j) {
            int r = rB[j];
            off_l += 0.5f * (fabsf(po[r * 2 + 0] - go[j * 2 + 0]) +
                             fabsf(po[r * 2 + 1] - go[j * 2 + 1]));
            float s4 = 0.f;
            #pragma unroll
            for (int k = 0; k < 4; ++k) s4 += fabsf(psz[r * 4 + k] - gsz[j * 4 + k]);
            size_l += 0.25f * s4;
        }
        float pose_l = 0.f;
        for (int j = 0; j < NG; ++j) {
            int r = rP[j];
            for (int c = 0; c < PDIM; ++c) {
                float d = ppL[r][c] - gpL[j][c];
                pose_l += d * d;
            }
        }
        ws[WS_OFF + bt] = off_l;
        ws[WS_SIZE + bt] = size_l;
        ws[WS_POSE + bt] = pose_l;
    }
}

// ---------------- Kernel C: deterministic finalize ----------------
__device__ __forceinline__ float block_sum(float v, float* red) {
    const int t = threadIdx.x;
    red[t] = v;
    __syncthreads();
    for (int s = 128; s > 0; s >>= 1) {
        if (t < s) red[t] += red[t + s];
        __syncthreads();
    }
    float r = red[0];
    __syncthreads();
    return r;
}

__global__ void finalize_kernel(const float* __restrict__ ws, float* __restrict__ out) {
    __shared__ float red[256];
    float s_mse = 0.f, s_off = 0.f, s_size = 0.f, s_pose = 0.f;
    for (int i = threadIdx.x; i < MSE_BLOCKS; i += 256) s_mse += ws[WS_MSE + i];
    for (int i = threadIdx.x; i < BT; i += 256) {
        s_off += ws[WS_OFF + i];
        s_size += ws[WS_SIZE + i];
        s_pose += ws[WS_POSE + i];
    }
    float mse_t = block_sum(s_mse, red);
    float off_t = block_sum(s_off, red);
    float size_t_ = block_sum(s_size, red);
    float pose_t = block_sum(s_pose, red);
    if (threadIdx.x == 0) {
        float center = mse_t / (float)MSE_N;
        float det = center + off_t / (float)BT + size_t_ / (float)BT;
        float pose = pose_t / (float)(BT * NP * NJ);
        out[0] = det + pose; // GAMMA = 1.0
    }
}

extern "C" void kernel_launch(void* const* d_in, const int* in_sizes, int n_in,
                              void* d_out, int out_size, void* d_ws, size_t ws_size,
                              hipStream_t stream) {
    (void)in_sizes; (void)n_in; (void)out_size; (void)ws_size;
    const float* hor_heatmap = (const float*)d_in[0];
    const float* hor_offset  = (const float*)d_in[1];
    const float* hor_bsize   = (const float*)d_in[2];
    const float* hor_center  = (const float*)d_in[3];
    const float* scores      = (const float*)d_in[4];
    const float* x_pose3d    = (const float*)d_in[5];
    const float* gt_heatmap  = (const float*)d_in[6];
    const float* gt_wh       = (const float*)d_in[7];
    const float* gt_off      = (const float*)d_in[8];
    const float* gt_center   = (const float*)d_in[9];
    const float* gt_pose     = (const float*)d_in[10];
    float* ws = (float*)d_ws;
    float* out = (float*)d_out;

    mse_partial_kernel<<<MSE_BLOCKS, MSE_THREADS, 0, stream>>>(hor_heatmap, gt_heatmap, ws);
    hungarian_kernel<<<BT, 32, 0, stream>>>(hor_offset, hor_bsize, hor_center, scores,
                                            x_pose3d, gt_wh, gt_off, gt_center, gt_pose, ws);
    finalize_kernel<<<1, 256, 0, stream>>>(ws, out);
}